// VectorQuantiser_22625887715902
// MI455X (gfx1250) — compile-verified
//
#include <hip/hip_runtime.h>
#include <hip/hip_bf16.h>

typedef __attribute__((ext_vector_type(2))) float v2f;
typedef __attribute__((ext_vector_type(8))) float v8f;

#define VQ_D        64      // embed dim
#define VQ_K        512     // codebook size
#define VQ_KC       256     // codebook columns per LDS chunk
#define VQ_STRIDE   68      // padded LDS row stride (floats) -> conflict-free B frags
#define VQ_ROWS_WG  128     // rows per workgroup (8 waves * 16 rows)

// LDS: transposed codebook chunk [256][68] + column norms [256]
__shared__ float s_E[VQ_KC * VQ_STRIDE];
__shared__ float s_norm[VQ_KC];

__global__ __launch_bounds__(256)
void VectorQuantiser_22625887715902_kernel(const float* __restrict__ x,
                                           const float* __restrict__ emb,
                                           float* __restrict__ out) {
    const int tid  = threadIdx.x;
    const int lane = tid & 31;
    const int wave = tid >> 5;
    const int n16  = lane & 15;   // column-within-tile / row-within-tile selector
    const int h    = lane >> 4;   // half-wave selector

    const long rowBase = (long)blockIdx.x * VQ_ROWS_WG + wave * 16;

    // ---- A fragments: 16 rows x 64 (fp32), WMMA 16x16x4 A layout ----
    // lane l holds x[l%16][4c + 2*(l>>4)] and x[l%16][4c + 2*(l>>4) + 1]
    const float* xrow = x + (rowBase + n16) * VQ_D;
    v2f a[16];
#pragma unroll
    for (int c = 0; c < 16; ++c) {
        a[c] = *(const v2f*)(xrow + 4 * c + 2 * h);
    }

    float bestD[8];
    int   bestI[8];
#pragma unroll
    for (int v = 0; v < 8; ++v) { bestD[v] = 3.4e38f; bestI[v] = 0; }

    for (int chunk = 0; chunk < 2; ++chunk) {
        // ---- Stage codebook chunk into LDS, transposed, + column norms ----
        {
            const int k = chunk * VQ_KC + tid;       // 256 threads -> 256 columns
            const float* col = emb + k;              // emb[d*512 + k]
            float nacc = 0.0f;
#pragma unroll 8
            for (int d = 0; d < VQ_D; ++d) {
                float v = col[d * VQ_K];             // coalesced across threads
                s_E[tid * VQ_STRIDE + d] = v;
                nacc += v * v;
            }
            s_norm[tid] = nacc;
        }
        __syncthreads();

        // ---- 16 column tiles of 16 codebook entries each ----
#pragma unroll 1
        for (int kt = 0; kt < 16; ++kt) {
            v8f acc = {};
            // B frag (4x16 fp32): lane l holds E[k0 + l%16][dbase + 2*(l>>4)], +1
            const float* bbase = &s_E[(kt * 16 + n16) * VQ_STRIDE + 2 * h];
#pragma unroll
            for (int c = 0; c < 16; ++c) {
                v2f b = *(const v2f*)(bbase + 4 * c);
                acc = __builtin_amdgcn_wmma_f32_16x16x4_f32(
                    /*neg_a=*/false, a[c], /*neg_b=*/false, b,
                    /*c_mod=*/(short)0, acc, /*reuse_a=*/false, /*reuse_b=*/false);
            }
            const float nrm  = s_norm[kt * 16 + n16];
            const int   curk = chunk * VQ_KC + kt * 16 + n16;
#pragma unroll
            for (int v = 0; v < 8; ++v) {
                float dist = __builtin_fmaf(-2.0f, acc[v], nrm);
                bool better = dist < bestD[v];
                bestD[v] = better ? dist : bestD[v];
                bestI[v] = better ? curk : bestI[v];
            }
        }
        __syncthreads();   // before overwriting LDS with next chunk
    }

    // ---- Argmin reduction across the 16 lanes sharing a row (wave32) ----
#pragma unroll
    for (int v = 0; v < 8; ++v) {
        float d = bestD[v];
        int   i = bestI[v];
#pragma unroll
        for (int off = 8; off >= 1; off >>= 1) {
            float od = __shfl_xor(d, off, 32);
            int   oi = __shfl_xor(i, off, 32);
            bool take = (od < d) || (od == d && oi < i);  // first-index tiebreak
            d = take ? od : d;
            i = take ? oi : i;
        }
        bestD[v] = d;
        bestI[v] = i;
    }

    // ---- Emit quantized rows: gather winning column (L2-resident table),
    //      coalesced b64 stores ----
#pragma unroll 1
    for (int rr = 0; rr < 16; ++rr) {
        int idx = __shfl(bestI[rr & 7], (rr >> 3) * 16, 32);
        float e0 = emb[(2 * lane + 0) * VQ_K + idx];
        float e1 = emb[(2 * lane + 1) * VQ_K + idx];
        v2f o = { e0, e1 };
        *(v2f*)(out + (rowBase + rr) * VQ_D + 2 * lane) = o;
    }
}

extern "C" void kernel_launch(void* const* d_in, const int* in_sizes, int n_in,
                              void* d_out, int out_size, void* d_ws, size_t ws_size,
                              hipStream_t stream) {
    const float* x   = (const float*)d_in[0];   // [64*32*32, 64] fp32
    const float* emb = (const float*)d_in[1];   // [64, 512] fp32
    float* out = (float*)d_out;                 // [64*32*32, 64] fp32

    const int N = in_sizes[0] / VQ_D;           // 65536 rows
    const int grid = N / VQ_ROWS_WG;            // 512 workgroups

    VectorQuantiser_22625887715902_kernel<<<grid, 256, 0, stream>>>(x, emb, out);
}